// PointNetSetAbstractionAttn_49237505082100
// MI455X (gfx1250) — compile-verified
//
#include <hip/hip_runtime.h>

typedef __attribute__((ext_vector_type(2))) float v2f;
typedef __attribute__((ext_vector_type(8))) float v8f;

#define Bn 8
#define Nn 8192
#define S_ATTN 128
#define S_NONE 384
#define Sn 512
#define NSAMPLE 64
#define RADIUS 0.4f
#define D_FEAT 64
#define IN_CH 67
#define C0PAD 68

// Output layout (flat floats, reference return order):
//  new_xyz       [B][3][S]   -> [0, 12288)
//  new_points    [B][128][S] -> [12288, 536576)
//  new_attention [B][1][S]   -> [536576, 540672)
#define OUT_NP_OFF 12288
#define OUT_ATT_OFF 536576

__device__ __forceinline__ v8f wmma_f32(v2f a, v2f b, v8f c) {
  // D = A(16x4 f32) x B(4x16 f32) + C(16x16 f32)
  return __builtin_amdgcn_wmma_f32_16x16x4_f32(
      /*neg_a=*/false, a, /*neg_b=*/false, b,
      /*c_mod=*/(short)0, c, /*reuse_a=*/false, /*reuse_b=*/false);
}

// ---------------------------------------------------------------------------
// Kernel 1: farthest point sampling, one block per (batch, branch).
// branch 0: attn*xyz,  K=128, rows [0,128)
// branch 1: (1-a)*xyz, K=384, rows [128,512)
// Writes new_xyz (branch-weighted sampled coords) and new_attention directly.
// ---------------------------------------------------------------------------
__global__ void __launch_bounds__(1024) fps_kernel(const float* __restrict__ xyz,
                                                   const float* __restrict__ attn,
                                                   float* __restrict__ out) {
  const int b = blockIdx.x >> 1;
  const int branch = blockIdx.x & 1;
  const int K = branch ? S_NONE : S_ATTN;
  const int rowoff = branch ? S_ATTN : 0;
  const int tid = threadIdx.x;
  const int wid = tid >> 5, lane = tid & 31;

  const float* xb = xyz + (size_t)b * 3 * Nn;
  const float* ab = attn + (size_t)b * Nn;

  float px[8], py[8], pz[8], mind[8];
#pragma unroll
  for (int i = 0; i < 8; ++i) {
    int n = tid + i * 1024;
    float a = ab[n];
    float w = branch ? (1.0f - a) : a;
    px[i] = w * xb[n];
    py[i] = w * xb[Nn + n];
    pz[i] = w * xb[2 * Nn + n];
    mind[i] = 1e10f;
  }

  __shared__ float s_bd[32];
  __shared__ int s_bn[32];
  __shared__ float s_l[3];
  __shared__ int s_last;

  int last = 0;
  for (int k = 0; k < K; ++k) {
    if (tid == 0) {
      float a = ab[last];
      float w = branch ? (1.0f - a) : a;
      float lx = w * xb[last];
      float ly = w * xb[Nn + last];
      float lz = w * xb[2 * Nn + last];
      int row = rowoff + k;
      out[(size_t)b * 1536 + row] = lx;
      out[(size_t)b * 1536 + 512 + row] = ly;
      out[(size_t)b * 1536 + 1024 + row] = lz;
      out[OUT_ATT_OFF + (size_t)b * 512 + row] = a;
      s_l[0] = lx; s_l[1] = ly; s_l[2] = lz;
    }
    __syncthreads();
    if (k == K - 1) break;

    const float lx = s_l[0], ly = s_l[1], lz = s_l[2];
    float bd = -1.0f;
    int bn = Nn;
#pragma unroll
    for (int i = 0; i < 8; ++i) {
      float dx = px[i] - lx, dy = py[i] - ly, dz = pz[i] - lz;
      float d2 = dx * dx + dy * dy + dz * dz;
      mind[i] = fminf(mind[i], d2);
      if (mind[i] > bd) { bd = mind[i]; bn = tid + i * 1024; }  // first-max in i order
    }
    // wave32 argmax reduce, lowest-index tie-break (matches jnp.argmax)
    for (int off = 16; off > 0; off >>= 1) {
      float od = __shfl_down(bd, off, 32);
      int on = __shfl_down(bn, off, 32);
      if (od > bd || (od == bd && on < bn)) { bd = od; bn = on; }
    }
    if (lane == 0) { s_bd[wid] = bd; s_bn[wid] = bn; }
    __syncthreads();
    if (wid == 0) {
      bd = s_bd[lane];
      bn = s_bn[lane];
      for (int off = 16; off > 0; off >>= 1) {
        float od = __shfl_down(bd, off, 32);
        int on = __shfl_down(bn, off, 32);
        if (od > bd || (od == bd && on < bn)) { bd = od; bn = on; }
      }
      if (lane == 0) s_last = bn;
    }
    __syncthreads();
    last = s_last;
  }
}

// ---------------------------------------------------------------------------
// One GEMM layer on a per-query tile: out[k][o] = relu(sum_c in[k][c]*W[o][c]+b[o])
// in/out live in LDS with row stride C0PAD. Uses V_WMMA_F32_16X16X4_F32.
// A frag: lane l -> M = l%16, K = c + 2*(l/16) (+0/.y +1)
// B frag: lane l -> N = l%16, K = c + 2*(l/16) (+0/.y +1); B[k][n] = W[n][k]
// D frag: vgpr r, lane l -> M = r + 8*(l/16), N = l%16
// ---------------------------------------------------------------------------
template <int CINPAD, int WSTRIDE>
__device__ __forceinline__ void mlp_layer(const float* __restrict__ W,
                                          const float* __restrict__ bias,
                                          const float* sin_, float* sout,
                                          int cout, int lane) {
  const int hl = lane >> 4;
  const int ll = lane & 15;
  for (int ot = 0; ot < cout / 16; ++ot) {
    const float bv = bias[ot * 16 + ll];
    for (int rt = 0; rt < 4; ++rt) {
      v8f acc = {};
      const int m = rt * 16 + ll;
      const int o = ot * 16 + ll;
      for (int c = 0; c < CINPAD; c += 4) {
        const int kk = c + (hl << 1);
        v2f a, bf;
        a.x = sin_[m * C0PAD + kk];
        a.y = sin_[m * C0PAD + kk + 1];
        bf.x = W[o * WSTRIDE + kk];
        bf.y = W[o * WSTRIDE + kk + 1];
        acc = wmma_f32(a, bf, acc);
      }
#pragma unroll
      for (int r = 0; r < 8; ++r) {
        float v = acc[r] + bv;
        v = v > 0.0f ? v : 0.0f;
        const int krow = rt * 16 + r + 8 * hl;
        sout[krow * C0PAD + ot * 16 + ll] = v;
      }
    }
  }
}

// Layer 3 (64 -> 128) fused with max-pool over the 64 samples (rows).
__device__ __forceinline__ void mlp_layer3_max(const float* __restrict__ W,
                                               const float* __restrict__ bias,
                                               const float* sin_,
                                               float* __restrict__ out,
                                               int lane, int b, int s) {
  const int hl = lane >> 4;
  const int ll = lane & 15;
  for (int ot = 0; ot < 8; ++ot) {  // 128 output channels
    const float bv = bias[ot * 16 + ll];
    v8f mx = {};  // post-ReLU values are >= 0, so 0 is a safe identity
    for (int rt = 0; rt < 4; ++rt) {
      v8f acc = {};
      const int m = rt * 16 + ll;
      const int o = ot * 16 + ll;
      for (int c = 0; c < 64; c += 4) {
        const int kk = c + (hl << 1);
        v2f a, bf;
        a.x = sin_[m * C0PAD + kk];
        a.y = sin_[m * C0PAD + kk + 1];
        bf.x = W[o * 64 + kk];
        bf.y = W[o * 64 + kk + 1];
        acc = wmma_f32(a, bf, acc);
      }
#pragma unroll
      for (int r = 0; r < 8; ++r) {
        float v = acc[r] + bv;
        v = v > 0.0f ? v : 0.0f;
        mx[r] = fmaxf(mx[r], v);
      }
    }
    // reduce over rows: in-lane (8) then across lane halves (shfl_xor 16)
    float m8 = mx[0];
#pragma unroll
    for (int r = 1; r < 8; ++r) m8 = fmaxf(m8, mx[r]);
    float oth = __shfl_xor(m8, 16, 32);
    m8 = fmaxf(m8, oth);
    if (lane < 16) {
      const int o = ot * 16 + lane;
      out[OUT_NP_OFF + (size_t)b * 65536 + (size_t)o * 512 + s] = m8;
    }
  }
}

// ---------------------------------------------------------------------------
// Kernel 2: fused ball-query + grouping + 3-layer MLP + max-pool.
// One wave32 block per (b, s) query. LDS: padded W0 + feat + act ~ 52.5 KB.
// ---------------------------------------------------------------------------
__global__ void __launch_bounds__(32) sa_fused_kernel(
    const float* __restrict__ xyz, const float* __restrict__ pts,
    const float* __restrict__ w0, const float* __restrict__ b0,
    const float* __restrict__ w1, const float* __restrict__ b1,
    const float* __restrict__ w2, const float* __restrict__ b2,
    float* __restrict__ out) {
  __shared__ float s_w0[64 * C0PAD];
  __shared__ float s_feat[64 * C0PAD];
  __shared__ float s_act[64 * C0PAD];
  __shared__ int s_idx[NSAMPLE];

  const int lane = threadIdx.x;
  const int b = blockIdx.x >> 9;
  const int s = blockIdx.x & 511;

  // stage zero-padded W0 [64][68] into LDS
  for (int t = lane; t < 64 * C0PAD; t += 32) {
    int r = t / C0PAD, c = t - r * C0PAD;
    s_w0[t] = (c < IN_CH) ? w0[r * IN_CH + c] : 0.0f;
  }

  const float* xb = xyz + (size_t)b * 3 * Nn;
  const float qx = out[(size_t)b * 1536 + s];
  const float qy = out[(size_t)b * 1536 + 512 + s];
  const float qz = out[(size_t)b * 1536 + 1024 + s];

  // ball query: first NSAMPLE indices (ascending) with d2 < r^2
  const float R2 = RADIUS * RADIUS;
  int count = 0;
  for (int base = 0; base < Nn && count < NSAMPLE; base += 32) {
    __builtin_prefetch(xb + base + 256, 0, 0);  // global_prefetch_b8
    const int n = base + lane;
    const float dx = xb[n] - qx;
    const float dy = xb[Nn + n] - qy;
    const float dz = xb[2 * Nn + n] - qz;
    const float d2 = dx * dx + dy * dy + dz * dz;
    const bool hit = d2 < R2;
    const unsigned mask = (unsigned)__ballot(hit);
    const int slot = count + __popc(mask & ((1u << lane) - 1u));
    if (hit && slot < NSAMPLE) s_idx[slot] = n;
    count += __popc(mask);
  }
  __syncthreads();
  const int cnt = count < NSAMPLE ? count : NSAMPLE;

  // build feat [64][68]: ch 0..2 = grouped_xyz - q (pad slot: 0 - q),
  // ch 3..66 = points[ch-3][idx<0 ? N-1 : idx], ch 67 = 0 (K-pad)
  const float* pb = pts + (size_t)b * D_FEAT * Nn;
  for (int k = 0; k < NSAMPLE; ++k) {
    const int n = (k < cnt) ? s_idx[k] : -1;
    const int gi = (n >= 0) ? n : (Nn - 1);
#pragma unroll
    for (int j = 0; j < 3; ++j) {
      const int ch = lane + j * 32;
      if (ch < C0PAD) {
        float v;
        if (ch < 3) {
          float g = (n >= 0) ? xb[ch * Nn + n] : 0.0f;
          v = g - ((ch == 0) ? qx : (ch == 1) ? qy : qz);
        } else if (ch < IN_CH) {
          v = pb[(ch - 3) * Nn + gi];
        } else {
          v = 0.0f;
        }
        s_feat[k * C0PAD + ch] = v;
      }
    }
  }
  __syncthreads();

  // layer 1: feat[64][68] x W0^T -> act[64][64]
  mlp_layer<C0PAD, C0PAD>(s_w0, b0, s_feat, s_act, 64, lane);
  __syncthreads();
  // layer 2: act x W1^T -> feat buffer (reuse)
  mlp_layer<64, 64>(w1, b1, s_act, s_feat, 64, lane);
  __syncthreads();
  // layer 3 + max-pool -> new_points[b][:][s]
  mlp_layer3_max(w2, b2, s_feat, out, lane, b, s);
}

extern "C" void kernel_launch(void* const* d_in, const int* in_sizes, int n_in,
                              void* d_out, int out_size, void* d_ws, size_t ws_size,
                              hipStream_t stream) {
  const float* xyz = (const float*)d_in[0];
  const float* pts = (const float*)d_in[1];
  const float* attn = (const float*)d_in[2];
  const float* w0 = (const float*)d_in[3];
  const float* b0 = (const float*)d_in[4];
  const float* w1 = (const float*)d_in[5];
  const float* b1 = (const float*)d_in[6];
  const float* w2 = (const float*)d_in[7];
  const float* b2 = (const float*)d_in[8];
  float* out = (float*)d_out;

  // FPS (writes new_xyz + new_attention regions of out)
  hipLaunchKernelGGL(fps_kernel, dim3(Bn * 2), dim3(1024), 0, stream, xyz, attn, out);
  // fused ball-query + grouping + WMMA MLP + max-pool (writes new_points)
  hipLaunchKernelGGL(sa_fused_kernel, dim3(Bn * Sn), dim3(32), 0, stream,
                     xyz, pts, w0, b0, w1, b1, w2, b2, out);
}